// ResponseDecoder_48241072668869
// MI455X (gfx1250) — compile-verified
//
#include <hip/hip_runtime.h>
#include <math.h>

// ---------------------------------------------------------------------------
// CDNA5 (gfx1250) wave32 WMMA f32 fragments:
//   V_WMMA_F32_16X16X4_F32 : A 16x4 (2 VGPR/lane), B 4x16 (2 VGPR/lane),
//                            C/D 16x16 f32 (8 VGPR/lane)
//   A lane layout: lanes 0-15 -> M=lane, K = k+{0,1}; lanes 16-31 -> K = k+{2,3}
//   B mirrors A with N in place of M (row-major-W == B^T source).
//   C/D: VGPR r -> M = r + (lane>=16 ? 8 : 0), N = lane & 15.
// ---------------------------------------------------------------------------
typedef float v2f __attribute__((ext_vector_type(2)));
typedef float v4f __attribute__((ext_vector_type(4)));
typedef float v8f __attribute__((ext_vector_type(8)));

static constexpr int EPI_NONE   = 0;
static constexpr int EPI_BIAS   = 1;
static constexpr int EPI_TANH   = 2;
static constexpr int EPI_ROWADD = 4;

// C[m,n] = act( sum_k X[m,k]*W[n,k] (+ bias[n]) (+ rowadd[m % rowmod, n]) )
// Block = 128 threads = 4 wave32s in a 2x2 arrangement -> 32x32 C tile.
// Requires M%32==0 and N%32==0 (holds at every call site below).
template<int EPI>
__global__ __launch_bounds__(128)
void gemm_xwT_wmma(const float* __restrict__ X, int ldx,
                   const float* __restrict__ W, int ldw,
                   const float* __restrict__ bias,
                   const float* __restrict__ rowadd, int rowmod, int rowld,
                   float* __restrict__ C, int ldc,
                   int M, int N, int K)
{
    const int lane  = threadIdx.x & 31;
    const int wave  = threadIdx.x >> 5;
    const int mbase = blockIdx.y * 32 + (wave >> 1) * 16;
    const int nbase = blockIdx.x * 32 + (wave & 1) * 16;
    const int l15   = lane & 15;
    const int hi    = lane >> 4;     // 0 or 1
    const int koff  = hi * 2;

    const float* Xrow = X + (size_t)(mbase + l15) * ldx;  // A row m for this lane
    const float* Wrow = W + (size_t)(nbase + l15) * ldw;  // B col n for this lane

    v8f c = {};
    const int Kmain = K & ~3;
#pragma unroll 4
    for (int k = 0; k < Kmain; k += 4) {
        v2f a, b;
        a.x = Xrow[k + koff];
        a.y = Xrow[k + koff + 1];
        b.x = Wrow[k + koff];
        b.y = Wrow[k + koff + 1];
        c = __builtin_amdgcn_wmma_f32_16x16x4_f32(false, a, false, b,
                                                  (short)0, c, false, false);
    }
    // Scalar K remainder (only the K=1541 GRU-input GEMM takes this path).
    for (int k = Kmain; k < K; ++k) {
        float wv = Wrow[k];
#pragma unroll
        for (int r = 0; r < 8; ++r)
            c[r] += X[(size_t)(mbase + hi * 8 + r) * ldx + k] * wv;
    }
#pragma unroll
    for (int r = 0; r < 8; ++r) {
        int m = mbase + hi * 8 + r;
        int n = nbase + l15;
        float v = c[r];
        if (EPI & EPI_BIAS)   v += bias[n];
        if (EPI & EPI_ROWADD) v += rowadd[(size_t)(m % rowmod) * rowld + n];
        if (EPI & EPI_TANH)   v = tanhf(v);
        C[(size_t)m * ldc + n] = v;
    }
}

// out[(r%B)*T + r/B] = dot(X[r, 0:H], vvec[(r%vmod)*vld : +H]); one wave per row.
// H % 128 == 0; X rows and vvec rows 16B-aligned -> b128 loads.
__global__ __launch_bounds__(256)
void rowdot(const float* __restrict__ X, int ldx,
            const float* __restrict__ vvec, int vmod, int vld,
            float* __restrict__ out, int B, int T, int H, int R)
{
    int r = blockIdx.x * (blockDim.x >> 5) + (threadIdx.x >> 5);
    if (r >= R) return;
    int lane = threadIdx.x & 31;
    const v4f* x = (const v4f*)(X + (size_t)r * ldx);
    const v4f* v = (const v4f*)(vvec + (size_t)(r % vmod) * vld);
    int H4 = H >> 2;
    float acc = 0.f;
    for (int h = lane; h < H4; h += 32) {
        v4f xv = x[h], vv = v[h];
        acc += xv.x * vv.x + xv.y * vv.y + xv.z * vv.z + xv.w * vv.w;
    }
    for (int off = 16; off; off >>= 1) acc += __shfl_down(acc, off, 32);
    if (lane == 0) out[(size_t)(r % B) * T + (r / B)] = acc;
}

// In-place softmax along T (T<=128); one wave per row.
__global__ __launch_bounds__(256)
void softmax_rows(float* __restrict__ s, int B, int T)
{
    int b = blockIdx.x * (blockDim.x >> 5) + (threadIdx.x >> 5);
    if (b >= B) return;
    int lane = threadIdx.x & 31;
    float* row = s + (size_t)b * T;
    float mx = -3.402823466e38f;
    for (int t = lane; t < T; t += 32) mx = fmaxf(mx, row[t]);
    for (int off = 16; off; off >>= 1) mx = fmaxf(mx, __shfl_xor(mx, off, 32));
    float sum = 0.f;
    for (int t = lane; t < T; t += 32) { float e = __expf(row[t] - mx); row[t] = e; sum += e; }
    for (int off = 16; off; off >>= 1) sum += __shfl_xor(sum, off, 32);
    float inv = 1.f / sum;
    for (int t = lane; t < T; t += 32) row[t] *= inv;
}

// ctx[b,h] = sum_t a[b,t] * enc[t,b,h]
__global__ __launch_bounds__(256)
void attn_ctx(const float* __restrict__ a, const float* __restrict__ enc,
              float* __restrict__ ctx, int B, int T, int H)
{
    int b = blockIdx.y;
    int h = blockIdx.x * blockDim.x + threadIdx.x;
    float acc = 0.f;
    for (int t = 0; t < T; ++t)
        acc += a[(size_t)b * T + t] * enc[((size_t)t * B + b) * H + h];
    ctx[(size_t)b * H + h] = acc;
}

// gru_in[b,:] = [ emb[m_t[b]] | u_ctx[b] | z_ctx[b] | degree[b] ]
__global__ __launch_bounds__(256)
void build_gru_in(const float* __restrict__ emb, const int* __restrict__ mt,
                  const float* __restrict__ uctx, const float* __restrict__ zctx,
                  const float* __restrict__ degree, float* __restrict__ out,
                  int B, int E, int H, int D)
{
    int gin = E + 2 * H + D;
    int idx = blockIdx.x * blockDim.x + threadIdx.x;
    if (idx >= B * gin) return;
    int b = idx / gin, j = idx - b * gin;
    float v;
    if (j < E)              v = emb[(size_t)mt[b] * E + j];
    else if (j < E + H)     v = uctx[(size_t)b * H + (j - E)];
    else if (j < E + 2 * H) v = zctx[(size_t)b * H + (j - E - H)];
    else                    v = degree[(size_t)b * D + (j - E - 2 * H)];
    out[idx] = v;
}

// gen_in[b,:] = [ z_ctx[b] | u_ctx[b] | h_new[b] ]
__global__ __launch_bounds__(256)
void build_gen_in(const float* __restrict__ z, const float* __restrict__ u,
                  const float* __restrict__ h, float* __restrict__ out, int B, int H)
{
    int idx = blockIdx.x * blockDim.x + threadIdx.x;
    if (idx >= B * 3 * H) return;
    int b = idx / (3 * H), j = idx - b * 3 * H;
    float v = (j < H) ? z[(size_t)b * H + j]
            : (j < 2 * H) ? u[(size_t)b * H + (j - H)]
            : h[(size_t)b * H + (j - 2 * H)];
    out[idx] = v;
}

// GRU gates; writes h_new to ws and to both gru_out output slots.
__global__ __launch_bounds__(256)
void gru_gate(const float* __restrict__ gi, const float* __restrict__ gh,
              const float* __restrict__ hprev, float* __restrict__ hnew,
              float* __restrict__ out1, float* __restrict__ out2, int B, int H)
{
    int idx = blockIdx.x * blockDim.x + threadIdx.x;
    if (idx >= B * H) return;
    int b = idx / H, j = idx - b * H;
    const float* gib = gi + (size_t)b * 3 * H;
    const float* ghb = gh + (size_t)b * 3 * H;
    float r = 1.f / (1.f + __expf(-(gib[j] + ghb[j])));
    float z = 1.f / (1.f + __expf(-(gib[H + j] + ghb[H + j])));
    float n = tanhf(gib[2 * H + j] + r * ghb[2 * H + j]);
    float h = (1.f - z) * n + z * hprev[idx];
    hnew[idx] = h; out1[idx] = h; out2[idx] = h;
}

// e[b,t] = exp(z_score - max); mrow[b] = max   (one wave per b, T=64)
__global__ __launch_bounds__(256)
void zscore_exp(float* __restrict__ zs, float* __restrict__ mrow, int B, int T)
{
    int b = blockIdx.x * (blockDim.x >> 5) + (threadIdx.x >> 5);
    if (b >= B) return;
    int lane = threadIdx.x & 31;
    float* row = zs + (size_t)b * T;
    float mx = -3.402823466e38f;
    for (int t = lane; t < T; t += 32) mx = fmaxf(mx, row[t]);
    for (int off = 16; off; off >>= 1) mx = fmaxf(mx, __shfl_xor(mx, off, 32));
    for (int t = lane; t < T; t += 32) row[t] = __expf(row[t] - mx);
    if (lane == 0) mrow[b] = mx;
}

// big[b, V+vp..vp+3] = log( sum_t e[b,t]*sparse[b,t,vp..vp+3] ) + mrow[b]
// 264 MB single-use stream: 16B non-temporal loads (b128, TH=NT) so the
// single pass runs at HBM rate without evicting the L2-resident weights.
__global__ __launch_bounds__(256)
void copy_logits(const float* __restrict__ e, const float* __restrict__ mrow,
                 const float* __restrict__ sparse, float* __restrict__ big,
                 int T, int VT, int NC, int V)
{
    __shared__ float es[64];
    int b = blockIdx.y;
    if (threadIdx.x < T) es[threadIdx.x] = e[(size_t)b * T + threadIdx.x];
    __syncthreads();
    int vp = (blockIdx.x * blockDim.x + threadIdx.x) * 4;
    if (vp >= VT) return;
    const v4f* sp = (const v4f*)(sparse + (size_t)b * T * VT + vp);
    size_t strideq = (size_t)(VT >> 2);   // row stride in v4f units
    v4f acc = {0.f, 0.f, 0.f, 0.f};
    for (int t = 0; t < T; ++t) {
        v4f sv = __builtin_nontemporal_load(sp + (size_t)t * strideq);
        float ev = es[t];
        acc.x += ev * sv.x; acc.y += ev * sv.y;
        acc.z += ev * sv.z; acc.w += ev * sv.w;
    }
    float m = mrow[b];
    float* dst = big + (size_t)b * NC + V + vp;
    dst[0] = __logf(acc.x) + m;
    dst[1] = __logf(acc.y) + m;
    dst[2] = __logf(acc.z) + m;
    dst[3] = __logf(acc.w) + m;
}

// softmax over 16064 logits + combine: proba[b,j<V]=gen+cp, proba[b,j>=V]=cp
__global__ __launch_bounds__(256)
void final_softmax(const float* __restrict__ big, float* __restrict__ proba,
                   int NC, int V, int VT)
{
    __shared__ float sh[8];
    __shared__ float bval[2];
    int b = blockIdx.x;
    const float* row = big + (size_t)b * NC;
    int lane = threadIdx.x & 31, wid = threadIdx.x >> 5;

    float mx = -3.402823466e38f;
    for (int i = threadIdx.x; i < NC; i += blockDim.x) mx = fmaxf(mx, row[i]);
    for (int off = 16; off; off >>= 1) mx = fmaxf(mx, __shfl_xor(mx, off, 32));
    if (lane == 0) sh[wid] = mx;
    __syncthreads();
    if (threadIdx.x == 0) {
        float m = sh[0];
        for (int w = 1; w < 8; ++w) m = fmaxf(m, sh[w]);
        bval[0] = m;
    }
    __syncthreads();
    mx = bval[0];

    float s = 0.f;
    for (int i = threadIdx.x; i < NC; i += blockDim.x) s += __expf(row[i] - mx);
    for (int off = 16; off; off >>= 1) s += __shfl_xor(s, off, 32);
    if (lane == 0) sh[wid] = s;
    __syncthreads();
    if (threadIdx.x == 0) {
        float t = 0.f;
        for (int w = 0; w < 8; ++w) t += sh[w];
        bval[1] = t;
    }
    __syncthreads();
    float inv = 1.f / bval[1];

    float* pb = proba + (size_t)b * VT;
    for (int j = threadIdx.x; j < VT; j += blockDim.x) {
        float cp = __expf(row[V + j] - mx) * inv;
        float g  = (j < V) ? __expf(row[j] - mx) * inv : 0.f;
        pb[j] = g + cp;
    }
}

extern "C" void kernel_launch(void* const* d_in, const int* in_sizes, int n_in,
                              void* d_out, int out_size, void* d_ws, size_t ws_size,
                              hipStream_t stream)
{
    (void)in_sizes; (void)n_in; (void)out_size; (void)ws_size;
    const int B = 128, Tz = 64, Tu = 128, H = 512, E = 512, D = 5, V = 8000;
    const int G3  = 3 * H;          // 1536
    const int GIN = E + 2 * H + D;  // 1541
    const int VT  = V + Tz;         // 8064
    const int NC  = V + VT;         // 16064

    const float* z_enc  = (const float*)d_in[0];
    const float* u_enc  = (const float*)d_in[1];
    const int*   m_t    = (const int*)  d_in[2];
    const float* degree = (const float*)d_in[3];
    const float* hidden = (const float*)d_in[4];
    const float* sparse = (const float*)d_in[5];
    const float* emb    = (const float*)d_in[6];
    const float* Wz     = (const float*)d_in[7];
    const float* bz     = (const float*)d_in[8];
    const float* vz     = (const float*)d_in[9];
    const float* Wu     = (const float*)d_in[10];
    const float* bu     = (const float*)d_in[11];
    const float* vu     = (const float*)d_in[12];
    const float* Wih    = (const float*)d_in[13];
    const float* Whh    = (const float*)d_in[14];
    const float* bih    = (const float*)d_in[15];
    const float* bhh    = (const float*)d_in[16];
    const float* Wp     = (const float*)d_in[17];
    const float* bp     = (const float*)d_in[18];
    const float* Wc2    = (const float*)d_in[19];
    const float* bc2    = (const float*)d_in[20];

    float* out   = (float*)d_out;
    float* proba = out;                         // (B, VT)
    float* gru1  = out + (size_t)B * VT;        // (1,B,H)
    float* gru2  = gru1 + (size_t)B * H;        // (1,B,H)

    float* ws = (float*)d_ws;
    size_t off = 0;
    auto alloc = [&](size_t n) { float* p = ws + off; off += n; return p; };
    float* hz   = alloc((size_t)B * H);
    float* hu   = alloc((size_t)B * H);
    float* ez   = alloc((size_t)Tz * B * H);   // energy_z; later reused as cs
    float* eu   = alloc((size_t)Tu * B * H);   // energy_u; later reused as big logits
    float* sz   = alloc((size_t)B * Tz);
    float* su   = alloc((size_t)B * Tu);
    float* zc   = alloc((size_t)B * H);
    float* uc   = alloc((size_t)B * H);
    float* gin  = alloc((size_t)B * GIN);
    float* gi   = alloc((size_t)B * G3);
    float* gh   = alloc((size_t)B * G3);
    float* hn   = alloc((size_t)B * H);
    float* gen  = alloc((size_t)B * G3);
    float* zsc  = alloc((size_t)B * Tz);
    float* mrow = alloc((size_t)B);
    float* cs   = ez;   // alias: free after attention scores
    float* big  = eu;   // alias: free after attention scores; (B, NC)

    dim3 blk128(128), blk256(256);

    // 1) hidden @ W1^T (W1 = W[:, :H], row stride 2H) for both attentions
    gemm_xwT_wmma<EPI_NONE><<<dim3(H / 32, B / 32), blk128, 0, stream>>>(
        hidden, H, Wz, 2 * H, nullptr, nullptr, 1, 0, hz, H, B, H, H);
    gemm_xwT_wmma<EPI_NONE><<<dim3(H / 32, B / 32), blk128, 0, stream>>>(
        hidden, H, Wu, 2 * H, nullptr, nullptr, 1, 0, hu, H, B, H, H);

    // 2) energy = tanh(enc @ W2^T + b + h_part[b])  (rows flattened t*B+b)
    gemm_xwT_wmma<EPI_BIAS | EPI_ROWADD | EPI_TANH>
        <<<dim3(H / 32, Tz * B / 32), blk128, 0, stream>>>(
        z_enc, H, Wz + H, 2 * H, bz, hz, B, H, ez, H, Tz * B, H, H);
    gemm_xwT_wmma<EPI_BIAS | EPI_ROWADD | EPI_TANH>
        <<<dim3(H / 32, Tu * B / 32), blk128, 0, stream>>>(
        u_enc, H, Wu + H, 2 * H, bu, hu, B, H, eu, H, Tu * B, H, H);

    // 3) scores[b,t] = energy . v
    rowdot<<<dim3((Tz * B + 7) / 8), blk256, 0, stream>>>(ez, H, vz, 1, 0, sz, B, Tz, H, Tz * B);
    rowdot<<<dim3((Tu * B + 7) / 8), blk256, 0, stream>>>(eu, H, vu, 1, 0, su, B, Tu, H, Tu * B);

    // 4) softmax over t
    softmax_rows<<<dim3(B / 8), blk256, 0, stream>>>(sz, B, Tz);
    softmax_rows<<<dim3(B / 8), blk256, 0, stream>>>(su, B, Tu);

    // 5) contexts
    attn_ctx<<<dim3(H / 256, B), blk256, 0, stream>>>(sz, z_enc, zc, B, Tz, H);
    attn_ctx<<<dim3(H / 256, B), blk256, 0, stream>>>(su, u_enc, uc, B, Tu, H);

    // 6) GRU
    build_gru_in<<<dim3((B * GIN + 255) / 256), blk256, 0, stream>>>(
        emb, m_t, uc, zc, degree, gin, B, E, H, D);
    gemm_xwT_wmma<EPI_BIAS><<<dim3(G3 / 32, B / 32), blk128, 0, stream>>>(
        gin, GIN, Wih, GIN, bih, nullptr, 1, 0, gi, G3, B, G3, GIN);
    gemm_xwT_wmma<EPI_BIAS><<<dim3(G3 / 32, B / 32), blk128, 0, stream>>>(
        hidden, H, Whh, H, bhh, nullptr, 1, 0, gh, G3, B, G3, H);
    gru_gate<<<dim3(B * H / 256), blk256, 0, stream>>>(gi, gh, hidden, hn, gru1, gru2, B, H);

    // 7) generation logits into big[:, :V]
    build_gen_in<<<dim3(B * G3 / 256), blk256, 0, stream>>>(zc, uc, hn, gen, B, H);
    gemm_xwT_wmma<EPI_BIAS><<<dim3(V / 32, B / 32), blk128, 0, stream>>>(
        gen, G3, Wp, G3, bp, nullptr, 1, 0, big, NC, B, V, G3);

    // 8) copy scores: cs = tanh(z_enc @ Wc2^T + bc2); z_score = cs . h_new[b]
    gemm_xwT_wmma<EPI_BIAS | EPI_TANH>
        <<<dim3(H / 32, Tz * B / 32), blk128, 0, stream>>>(
        z_enc, H, Wc2, H, bc2, nullptr, 1, 0, cs, H, Tz * B, H, H);
    rowdot<<<dim3((Tz * B + 7) / 8), blk256, 0, stream>>>(cs, H, hn, B, H, zsc, B, Tz, H, Tz * B);
    zscore_exp<<<dim3(B / 8), blk256, 0, stream>>>(zsc, mrow, B, Tz);

    // 9) z_copy = log(e @ sparse) + m  into big[:, V:]  (b128 NT stream, 264 MB)
    copy_logits<<<dim3((VT / 4 + 255) / 256, B), blk256, 0, stream>>>(
        zsc, mrow, sparse, big, Tz, VT, NC, V);

    // 10) final softmax over NC + combine into proba
    final_softmax<<<dim3(B), blk256, 0, stream>>>(big, proba, NC, V, VT);
}